// _ESM_57715770523839
// MI455X (gfx1250) — compile-verified
//
#include <hip/hip_runtime.h>
#include <stdint.h>

// ---- problem constants (from reference) ----
#define TT      512
#define BB      1024
#define NIN     28
#define NOUT    28
#define NW      457        // T - INPUT - OUTPUT + 1
#define SEASLEN 519        // (M+1) + (T-1) = 8 + 511
#define CIN     144        // 28 (ES) + 112 (X) + 4 (S)
#define ROW4    36         // 144/4 float4 per insample row

static constexpr size_t SZ_INS   = (size_t)NW * BB * CIN;    // 67,387,392
static constexpr size_t SZ_OUTS  = (size_t)NW * BB * NOUT;   // 13,103,104
static constexpr size_t SZ_LEV   = (size_t)BB * TT;          //    524,288
static constexpr size_t SZ_SEAS  = (size_t)BB * SEASLEN;     //    531,456
static constexpr size_t OFF_OUTS = SZ_INS;
static constexpr size_t OFF_LEV  = OFF_OUTS + SZ_OUTS;
static constexpr size_t OFF_SEAS = OFF_LEV + SZ_LEV;
static constexpr size_t OFF_MASK = OFF_SEAS + SZ_SEAS;

typedef uint32_t u32x4 __attribute__((ext_vector_type(4)));
typedef int      i32x4 __attribute__((ext_vector_type(4)));
typedef int      i32x8 __attribute__((ext_vector_type(8)));
typedef float    f32x4 __attribute__((ext_vector_type(4)));

// rcp + 1 Newton step: ~0.5 ulp divide without the full v_div sequence
__device__ inline float fast_div(float a, float b) {
    float r = __builtin_amdgcn_rcpf(b);
    r = fmaf(fmaf(-b, r, 1.0f), r, r);
    return a * r;
}

// low 32 bits of a flat pointer to LDS == byte offset within the workgroup's LDS
__device__ inline uint32_t lds_off(const void* p) {
    return (uint32_t)(uintptr_t)p;
}

// TDM: 1D contiguous copy of n f32 elements global -> LDS.
// D# per CDNA5 ISA ch.8: group0 {count=1, lds_addr[63:32], global_addr[120:64], type=2},
// group1 {data_size=4B, tensor_dim0=n, tensor_dim1=1, tile_dim0=n, tile_dim1=1,
// dim0_stride=n}, groups 2/3 zero. This toolchain uses the 6-arg builtin form
// (u32x4, i32x8, i32x4, i32x4, i32x8, i32 cpol) — extra i32x8 zero-filled.
__device__ inline void tdm_copy_f32(uint32_t lds_addr, const float* gptr, uint32_t n) {
    uint64_t ga = (uint64_t)(uintptr_t)gptr;
    u32x4 g0;
    g0[0] = 1u;                                                  // count = 1
    g0[1] = lds_addr;
    g0[2] = (uint32_t)ga;
    g0[3] = (uint32_t)((ga >> 32) & 0x01FFFFFFu) | (2u << 30);   // addr[56:32] | type=2
    i32x8 g1;
    g1[0] = (int)(2u << 16);                                     // data_size = 4 bytes
    g1[1] = (int)((n & 0xFFFFu) << 16);                          // tensor_dim0[15:0]
    g1[2] = (int)(((n >> 16) & 0xFFFFu) | (1u << 16));           // dim0[31:16] | tensor_dim1=1
    g1[3] = (int)((n & 0xFFFFu) << 16);                          // tile_dim0 = n
    g1[4] = 1;                                                   // tile_dim1 = 1, tile_dim2 = 0
    g1[5] = (int)n;                                              // tensor_dim0_stride lo32
    g1[6] = 0;
    g1[7] = 0;
    i32x4 z4 = {0, 0, 0, 0};
    i32x8 z8 = {0, 0, 0, 0, 0, 0, 0, 0};
    __builtin_amdgcn_tensor_load_to_lds(g0, g1, z4, z4, z8, 0);
}

// -------- Kernel 1: ES recurrence, one lane per series --------
__global__ __launch_bounds__(128) void es_scan_kernel(
    const float* __restrict__ Y, const int* __restrict__ idxs,
    const float* __restrict__ emb, float* __restrict__ levels,
    float* __restrict__ seas) {
    int b = blockIdx.x * blockDim.x + threadIdx.x;
    if (b >= BB) return;
    const float* e = emb + (size_t)idxs[b] * 9;
    float lev_sms  = 1.0f / (1.0f + expf(-e[0]));
    float seas_sms = 1.0f / (1.0f + expf(-e[1]));
    float init[7];
#pragma unroll
    for (int k = 0; k < 7; ++k) init[k] = expf(e[2 + k]);
    float s0 = init[0];
    const float* Yr = Y + (size_t)b * TT;
    float* Lr = levels + (size_t)b * TT;
    float* Sr = seas + (size_t)b * SEASLEN;
    float lev = fast_div(Yr[0], s0);
    Lr[0] = lev;
#pragma unroll
    for (int k = 0; k < 7; ++k) Sr[k] = init[k];
    Sr[7] = s0;
    float buf[7];
#pragma unroll
    for (int k = 0; k < 6; ++k) buf[k] = init[k + 1];
    buf[6] = s0;
    float omL = 1.0f - lev_sms, omS = 1.0f - seas_sms;
    for (int t = 1; t < TT; ++t) {
        float y  = Yr[t];
        float st = buf[0];
        float nl = lev_sms * fast_div(y, st) + omL * lev;
        float ns = seas_sms * fast_div(y, nl) + omS * st;
        Lr[t]     = nl;
        Sr[7 + t] = ns;
#pragma unroll
        for (int k = 0; k < 6; ++k) buf[k] = buf[k + 1];
        buf[6] = ns;
        lev = nl;
    }
}

// -------- Kernel 2: windowed expansion, one block per series --------
__global__ __launch_bounds__(256) void windows_kernel(
    const float* __restrict__ S, const float* __restrict__ Y,
    const float* __restrict__ X, const float* __restrict__ mask,
    const float* __restrict__ levels, const float* __restrict__ seas,
    float* __restrict__ out_ins, float* __restrict__ out_outs,
    float* __restrict__ out_mask) {
    __shared__ float shY[TT];        // raw Y row (for outsample)
    __shared__ float shLogY[TT];     // log(Y)
    __shared__ float shX[2 * TT];    // X[b,0,:] ++ X[b,1,:]
    __shared__ float shM[TT];        // sample_mask row
    __shared__ float shL[TT];        // levels -> log(levels) in place
    __shared__ float shSe[520];      // seas   -> log(seas) in place (519 used)
    __shared__ float shS[4];         // static features

    const int b   = blockIdx.x;
    const int tid = threadIdx.x;

    if (tid == 0) {  // only wave 0 reaches this; TDM issues once per wave (EXEC ignored)
        tdm_copy_f32(lds_off(shY),  Y      + (size_t)b * TT,      TT);
        tdm_copy_f32(lds_off(shX),  X      + (size_t)b * 2 * TT,  2 * TT);
        tdm_copy_f32(lds_off(shM),  mask   + (size_t)b * TT,      TT);
        tdm_copy_f32(lds_off(shL),  levels + (size_t)b * TT,      TT);
        tdm_copy_f32(lds_off(shSe), seas   + (size_t)b * SEASLEN, SEASLEN);
        __builtin_amdgcn_s_wait_tensorcnt(0);
    }
    if (tid < 4) shS[tid] = S[(size_t)b * 4 + tid];
    __syncthreads();

    // precompute logs once per block: insample A = logY - logLev - logSeas
    for (int t = tid; t < TT; t += 256) {
        shLogY[t] = logf(shY[t]);
        shL[t]    = logf(shL[t]);
    }
    for (int t = tid; t < SEASLEN; t += 256) shSe[t] = logf(shSe[t]);
    __syncthreads();

    // ---- insample (W,B,144): 36 float4 per (w,b) row, fully coalesced NT stores ----
    for (int idx = tid; idx < NW * ROW4; idx += 256) {
        int w  = idx / ROW4;
        int k  = idx - w * ROW4;
        int c0 = k * 4;
        f32x4 v;
        if (c0 < NIN) {                       // log(Y/(lev_end*seas)) block
            float Lw = shL[w + NIN - 1];
            int base = w + c0;
            v.x = shLogY[base + 0] - Lw - shSe[base + 0];
            v.y = shLogY[base + 1] - Lw - shSe[base + 1];
            v.z = shLogY[base + 2] - Lw - shSe[base + 2];
            v.w = shLogY[base + 3] - Lw - shSe[base + 3];
        } else if (c0 < NIN + 112) {          // X windows: c = 28 + n*56 + j
            int cc = c0 - NIN;
            int n  = (cc >= 56) ? 1 : 0;
            int j  = cc - n * 56;
            const float* xs = &shX[n * TT + w + j];
            v.x = xs[0]; v.y = xs[1]; v.z = xs[2]; v.w = xs[3];
        } else {                              // static features broadcast
            v.x = shS[0]; v.y = shS[1]; v.z = shS[2]; v.w = shS[3];
        }
        f32x4* dst = (f32x4*)(out_ins + ((size_t)w * BB + b) * CIN);
        __builtin_nontemporal_store(v, dst + k);
    }

    // ---- outsample + mask (W,B,28): 7 float4 per row each ----
    for (int idx = tid; idx < NW * 7; idx += 256) {
        int w  = idx / 7;
        int k  = idx - w * 7;
        int i0 = k * 4;
        int base = w + NIN + i0;
        f32x4 v;
        v.x = shY[base + 0]; v.y = shY[base + 1];
        v.z = shY[base + 2]; v.w = shY[base + 3];
        f32x4* dsto = (f32x4*)(out_outs + ((size_t)w * BB + b) * NOUT);
        __builtin_nontemporal_store(v, dsto + k);
        f32x4 m;                               // mask index = 28 + w + i
        m.x = shM[base + 0]; m.y = shM[base + 1];
        m.z = shM[base + 2]; m.w = shM[base + 3];
        f32x4* dstm = (f32x4*)(out_mask + ((size_t)w * BB + b) * NOUT);
        __builtin_nontemporal_store(m, dstm + k);
    }
}

extern "C" void kernel_launch(void* const* d_in, const int* in_sizes, int n_in,
                              void* d_out, int out_size, void* d_ws, size_t ws_size,
                              hipStream_t stream) {
    const float* S    = (const float*)d_in[0];
    const float* Y    = (const float*)d_in[1];
    const float* X    = (const float*)d_in[2];
    const int*   idxs = (const int*)d_in[3];
    const float* mask = (const float*)d_in[4];
    const float* emb  = (const float*)d_in[5];
    float* out = (float*)d_out;

    float* out_ins  = out;
    float* out_outs = out + OFF_OUTS;
    float* out_lev  = out + OFF_LEV;
    float* out_seas = out + OFF_SEAS;
    float* out_mask = out + OFF_MASK;

    // Kernel 1: ES scan writes levels/seasonalities directly into their output slots.
    es_scan_kernel<<<BB / 128, 128, 0, stream>>>(Y, idxs, emb, out_lev, out_seas);
    // Kernel 2: one block per series; TDM-staged LDS, streaming float4 stores.
    windows_kernel<<<BB, 256, 0, stream>>>(S, Y, X, mask, out_lev, out_seas,
                                           out_ins, out_outs, out_mask);
}